// ChamferLoss_52544629899782
// MI455X (gfx1250) — compile-verified
//
#include <hip/hip_runtime.h>
#include <math.h>

typedef __attribute__((ext_vector_type(2))) float v2f;
typedef __attribute__((ext_vector_type(8))) float v8f;

#define EPSF 1e-12f
#define NPTS 512
#define TILE 16
#define NTILES (NPTS / TILE)   // 32
#define WG_THREADS 256
#define WAVES (WG_THREADS / 32)

__global__ void chamfer_zero_kernel(float* out) { out[0] = 0.0f; }

__global__ void __launch_bounds__(WG_THREADS)
chamfer_wmma_kernel(const float* __restrict__ p, const float* __restrict__ q,
                    float* __restrict__ out)
{
    // Per-batch staging in LDS. q-side stored with Minkowski signs (+,-,-,-) applied,
    // so every WMMA is a plain dot product.
    __shared__ float sPre[NPTS * 4];
    __shared__ float sPim[NPTS * 4];
    __shared__ float sQre[NPTS * 4];   // eta * q_re
    __shared__ float sQim[NPTS * 4];   // eta * q_im
    __shared__ float sArr[NPTS];       // M(p_re,p_re)
    __shared__ float sAri[NPTS];       // M(p_re,p_im)
    __shared__ float sBrr[NPTS];       // M(q_re,q_re)
    __shared__ float sBri[NPTS];       // M(q_re,q_im)
    __shared__ unsigned int sRowMin[NPTS];  // min over m of r2, per n (float bits)
    __shared__ unsigned int sColMin[NPTS];  // min over n of r2, per m (float bits)
    __shared__ float sRed[WG_THREADS];

    const int b   = blockIdx.x;
    const int gb  = gridDim.x;          // number of batches (64)
    const int tid = threadIdx.x;

    const float4* pre4 = (const float4*)p + (size_t)b * NPTS;
    const float4* pim4 = (const float4*)p + ((size_t)gb + b) * NPTS;
    const float4* qre4 = (const float4*)q + (size_t)b * NPTS;
    const float4* qim4 = (const float4*)q + ((size_t)gb + b) * NPTS;

    // ---- load + norm precompute phase (512 points each side, 2 per thread) ----
    for (int i = tid; i < NPTS; i += WG_THREADS) {
        float4 re = pre4[i];
        float4 im = pim4[i];
        sArr[i] = re.x * re.x - re.y * re.y - re.z * re.z - re.w * re.w;
        sAri[i] = re.x * im.x - re.y * im.y - re.z * im.z - re.w * im.w;
        ((float4*)sPre)[i] = re;
        ((float4*)sPim)[i] = im;

        float4 qr = qre4[i];
        float4 qi = qim4[i];
        sBrr[i] = qr.x * qr.x - qr.y * qr.y - qr.z * qr.z - qr.w * qr.w;
        sBri[i] = qr.x * qi.x - qr.y * qi.y - qr.z * qi.z - qr.w * qi.w;
        float4 eqr = make_float4(qr.x, -qr.y, -qr.z, -qr.w);
        float4 eqi = make_float4(qi.x, -qi.y, -qi.z, -qi.w);
        ((float4*)sQre)[i] = eqr;
        ((float4*)sQim)[i] = eqi;

        sRowMin[i] = 0x7F800000u;   // +inf
        sColMin[i] = 0x7F800000u;
    }
    __syncthreads();

    // ---- WMMA tile phase: each wave owns n-strips, sweeps all m-tiles ----
    const int wave = tid >> 5;
    const int lane = tid & 31;
    const int half = lane >> 4;        // K-pair selector: lanes 0-15 -> K{0,1}, 16-31 -> K{2,3}
    const int lrow = lane & 15;        // row (A) / column (B) within tile
    const int kofs = 2 * half;

    for (int nt = wave; nt < NTILES; nt += WAVES) {
        const int n0 = nt * TILE;

        // A operands (loop-invariant across m-tiles)
        const v2f a_re = *(const v2f*)&sPre[(n0 + lrow) * 4 + kofs];
        const v2f a_im = *(const v2f*)&sPim[(n0 + lrow) * 4 + kofs];

        // per-accumulator-row norms (row = n0 + r + 8*half)
        float arr8[8], ari8[8], rmin[8];
        #pragma unroll
        for (int r = 0; r < 8; ++r) {
            arr8[r] = sArr[n0 + r + 8 * half];
            ari8[r] = sAri[n0 + r + 8 * half];
            rmin[r] = INFINITY;
        }

        for (int mt = 0; mt < NTILES; ++mt) {
            const int m0 = mt * TILE;
            const v2f b_re = *(const v2f*)&sQre[(m0 + lrow) * 4 + kofs];
            const v2f b_im = *(const v2f*)&sQim[(m0 + lrow) * 4 + kofs];

            v8f acc1 = {};   // M(p_re, q_re)
            v8f acc2 = {};   // M(p_re, q_im) + M(p_im, q_re)
            acc1 = __builtin_amdgcn_wmma_f32_16x16x4_f32(
                false, a_re, false, b_re, (short)0, acc1, false, false);
            acc2 = __builtin_amdgcn_wmma_f32_16x16x4_f32(
                false, a_re, false, b_im, (short)0, acc2, false, false);
            acc2 = __builtin_amdgcn_wmma_f32_16x16x4_f32(
                false, a_im, false, b_re, (short)0, acc2, false, false);

            const float brr = sBrr[m0 + lrow];
            const float bri = sBri[m0 + lrow];
            float cmin = INFINITY;
            #pragma unroll
            for (int r = 0; r < 8; ++r) {
                float mr = arr8[r] + brr - 2.0f * acc1[r];   // m_real
                float mi = ari8[r] + bri - acc2[r];          // m_im / 2
                float r2 = mr * mr + 4.0f * mi * mi;         // m_real^2 + m_im^2
                cmin    = fminf(cmin, r2);
                rmin[r] = fminf(rmin[r], r2);
            }
            // partial column min (this lane covered 8 rows of this n-tile)
            atomicMin(&sColMin[m0 + lrow], __float_as_uint(cmin));
        }
        // flush row mins (this lane covered columns lrow mod 16 over all m-tiles)
        #pragma unroll
        for (int r = 0; r < 8; ++r)
            atomicMin(&sRowMin[n0 + r + 8 * half], __float_as_uint(rmin[r]));
    }
    __syncthreads();

    // ---- per-batch reduction: sum sqrt(min + EPS), then one global atomic ----
    float partial = 0.0f;
    for (int i = tid; i < NPTS; i += WG_THREADS) {
        partial += sqrtf(__uint_as_float(sRowMin[i]) + EPSF);
        partial += sqrtf(__uint_as_float(sColMin[i]) + EPSF);
    }
    sRed[tid] = partial;
    __syncthreads();
    for (int s = WG_THREADS / 2; s > 0; s >>= 1) {
        if (tid < s) sRed[tid] += sRed[tid + s];
        __syncthreads();
    }
    if (tid == 0) atomicAdd(out, sRed[0]);
}

extern "C" void kernel_launch(void* const* d_in, const int* in_sizes, int n_in,
                              void* d_out, int out_size, void* d_ws, size_t ws_size,
                              hipStream_t stream) {
    const float* p = (const float*)d_in[0];   // (2, B, 512, 4) f32
    const float* q = (const float*)d_in[1];   // (2, B, 512, 4) f32
    float* out = (float*)d_out;               // scalar f32

    const int B = in_sizes[0] / (2 * NPTS * 4);   // 64

    chamfer_zero_kernel<<<1, 1, 0, stream>>>(out);
    chamfer_wmma_kernel<<<B, WG_THREADS, 0, stream>>>(p, q, out);
}